// MatchingNet_82506321756930
// MI455X (gfx1250) — compile-verified
//
#include <hip/hip_runtime.h>
#include <math.h>

typedef float v2f __attribute__((ext_vector_type(2)));
typedef float v8f __attribute__((ext_vector_type(8)));

__device__ __forceinline__ v8f wmma4(v2f a, v2f b, v8f c) {
  // D = A(16x4 f32) * B(4x16 f32) + C(16x16 f32)
  return __builtin_amdgcn_wmma_f32_16x16x4_f32(false, a, false, b, (short)0, c, false, false);
}

// ---- model constants ----
constexpr int Bc   = 2;
constexpr int Lc   = 2304;
constexpr int Ntot = Bc * Lc;        // 4608
constexpr int SEQ  = 2 * Ntot;       // 9216
constexpr int DcK  = 256;
constexpr int DF   = 128;
constexpr int WWc  = 25;
constexpr int HF   = 192;
constexpr int PIX  = HF * HF;        // 36864
constexpr int RST  = SEQ * WWc;      // 230400 state rows

// ---- workspace layout (float offsets) ----
constexpr size_t OFF_STATE = 0;
constexpr size_t SZ_STATE  = (size_t)RST * DF;          // 29,491,200
constexpr size_t OFF_F     = OFF_STATE + SZ_STATE;
constexpr size_t SZ_F      = (size_t)4 * PIX * DF;      // 18,874,368
constexpr size_t OFF_SIM   = OFF_F + SZ_F;
constexpr size_t SZ_SIM    = (size_t)Bc * Lc * Lc;      // 10,616,832
constexpr size_t OFF_CD    = OFF_SIM + SZ_SIM;
constexpr size_t SZ_CD     = (size_t)SEQ * DcK;
constexpr size_t OFF_CEN   = OFF_CD + SZ_CD;
constexpr size_t SZ_CEN    = (size_t)SEQ * DF;
constexpr size_t OFF_CMG   = OFF_CEN + SZ_CEN;
constexpr size_t OFF_PE    = OFF_CMG + SZ_CEN;
constexpr size_t OFF_LRS   = OFF_PE + (size_t)WWc * DF;
constexpr size_t OFF_LCS   = OFF_LRS + Ntot;
constexpr size_t OFF_CMX   = OFF_LCS + Ntot;
constexpr size_t OFF_MSK   = OFF_CMX + Ntot;
constexpr size_t OFF_J     = OFF_MSK + Ntot;            // int
constexpr size_t OFF_ROF   = OFF_J + Ntot;              // int, 230400
constexpr size_t OFF_RO0   = OFF_ROF + RST;             // int, 115200
constexpr size_t OFF_RO1   = OFF_RO0 + (size_t)Ntot*WWc;// int, 115200
constexpr size_t OFF_FEAT1 = OFF_F;                     // reuse (needs 14,745,600 <= SZ_F)
constexpr size_t OFF_FEATV = OFF_SIM;                   // reuse (needs 589,824 <= SZ_SIM)

// =====================================================================
// Generic wave-tile WMMA GEMM: each wave computes a 16x64 tile of
// C = scale * A(MxK) * B(KxN) [+ bias[col]] [+ merge epilogue]
// A reader supports: TRANSA (A[m,k]=b0[k*lda0+m]), row-indirection
// (ro0[row] byte.. float offset; <0 => zero row) and 2-source concat.
// B reader supports TRANSB (B[k,n]=W[n*ldw+k]).
// =====================================================================
template<bool TA, bool TB>
__global__ __launch_bounds__(256) void gemm_k(
    const float* __restrict__ b0, const int* __restrict__ ro0, int lda0,
    const float* __restrict__ b1, const int* __restrict__ ro1, int split,
    const float* __restrict__ W, int ldw,
    const float* __restrict__ bias,
    float* __restrict__ C, int ldc,
    int M, int Nn, int K, float scale, int epi,
    const float* __restrict__ ecmg, const float* __restrict__ epe,
    long bsA, long bsB, long bsC)
{
  const int z = blockIdx.y;
  b0 += (size_t)z * bsA;
  W  += (size_t)z * bsB;
  C  += (size_t)z * bsC;
  const int Mt = M >> 4, Nt = Nn >> 6;
  const int wav = blockIdx.x * 8 + (threadIdx.x >> 5);
  if (wav >= Mt * Nt) return;
  const int mt = wav / Nt, nt = wav % Nt;
  const int row0 = mt << 4, col0 = nt << 6;
  const int lane = threadIdx.x & 31, m = lane & 15, h = lane >> 4;

  v8f acc[4];
#pragma unroll
  for (int j = 0; j < 4; ++j) acc[j] = (v8f){0,0,0,0,0,0,0,0};

  const int arow = row0 + m;
  for (int kb = 0; kb < K; kb += 4) {
    const int k0 = kb + 2 * h;                 // k0 even; lanes16-31 carry K-half {2,3}
    v2f a;
    if (TA) {
      a.x = b0[(size_t)k0 * lda0 + arow];
      a.y = b0[(size_t)(k0 + 1) * lda0 + arow];
    } else {
      float ax, ay;
      if (k0 < split) {
        int off = ro0 ? ro0[arow] : arow * lda0;
        bool zr = (ro0 && off < 0);
        ax = zr ? 0.f : b0[(size_t)off + k0];
        ay = zr ? 0.f : b0[(size_t)off + k0 + 1];
      } else {
        int off = ro1 ? ro1[arow] : arow * lda0;
        bool zr = (ro1 && off < 0);
        ax = zr ? 0.f : b1[(size_t)off + k0 - split];
        ay = zr ? 0.f : b1[(size_t)off + k0 + 1 - split];
      }
      a.x = ax; a.y = ay;
    }
#pragma unroll
    for (int j = 0; j < 4; ++j) {
      const int n = col0 + j * 16 + m;
      v2f b;
      if (TB) {
        b.x = W[(size_t)n * ldw + k0];
        b.y = W[(size_t)n * ldw + k0 + 1];
      } else {
        b.x = W[(size_t)k0 * ldw + n];
        b.y = W[(size_t)(k0 + 1) * ldw + n];
      }
      acc[j] = wmma4(a, b, acc[j]);
    }
  }

#pragma unroll
  for (int j = 0; j < 4; ++j) {
    const int col = col0 + j * 16 + m;
    const float bv = bias ? bias[col] : 0.f;
#pragma unroll
    for (int r = 0; r < 8; ++r) {
      const int row = row0 + r + 8 * h;        // lanes0-15 -> rows r, lanes16-31 -> rows r+8
      float v = acc[j][r] * scale + bv;
      if (epi) v += ecmg[(row / 25) * DF + col] + epe[(row % 25) * DF + col];
      C[(size_t)row * ldc + col] = v;
    }
  }
}

// =====================================================================
// Matching: sim -> log row/col sums -> mutual-NN (j, mask)
// =====================================================================
__global__ __launch_bounds__(256) void rowsum_k(const float* __restrict__ sim,
                                                float* __restrict__ lrs) {
  const int row = blockIdx.x * 8 + (threadIdx.x >> 5);
  if (row >= Ntot) return;
  const float* p = sim + (size_t)row * Lc;
  const int lane = threadIdx.x & 31;
  float s = 0.f;
  for (int mcol = lane; mcol < Lc; mcol += 32) s += expf(p[mcol]);
  for (int o = 16; o; o >>= 1) s += __shfl_xor(s, o);
  if (lane == 0) lrs[row] = logf(s);
}

__global__ __launch_bounds__(256) void col_k(const float* __restrict__ sim,
                                             const float* __restrict__ lrs,
                                             float* __restrict__ lcs,
                                             float* __restrict__ cmax) {
  const int idx = blockIdx.x * 256 + threadIdx.x;
  if (idx >= Ntot) return;
  const int b = idx / Lc, mcol = idx % Lc;
  const float* p = sim + (size_t)b * Lc * Lc + mcol;
  const float* lr = lrs + b * Lc;
  float cs = 0.f, mx = -3.4e38f;
  for (int l = 0; l < Lc; ++l) {
    const float v = p[(size_t)l * Lc];
    cs += expf(v);
    const float u = 2.0f * v - lr[l];
    mx = fmaxf(mx, u);
  }
  lcs[idx] = logf(cs);
  cmax[idx] = mx;
}

__global__ __launch_bounds__(256) void match_k(const float* __restrict__ sim,
                                               const float* __restrict__ lrs,
                                               const float* __restrict__ lcs,
                                               const float* __restrict__ cmax,
                                               int* __restrict__ jarr,
                                               float* __restrict__ maskf) {
  const int row = blockIdx.x * 8 + (threadIdx.x >> 5);
  if (row >= Ntot) return;
  const int b = row / Lc;
  const float* p = sim + (size_t)row * Lc;
  const float* lc = lcs + b * Lc;
  const int lane = threadIdx.x & 31;
  float best = -3.4e38f; int bi = 0x7fffffff;
  for (int mcol = lane; mcol < Lc; mcol += 32) {
    const float v = 2.0f * p[mcol] - lc[mcol];
    if (v > best) { best = v; bi = mcol; }
  }
  for (int o = 16; o; o >>= 1) {
    const float ob = __shfl_xor(best, o);
    const int oi = __shfl_xor(bi, o);
    if (ob > best || (ob == best && oi < bi)) { best = ob; bi = oi; }
  }
  if (lane == 0) {
    const float e = 2.0f * p[bi] - lrs[row];
    jarr[row] = bi;
    maskf[row] = (e >= cmax[b * Lc + bi]) ? 1.f : 0.f;
  }
}

// =====================================================================
// Row-offset builders, center gather, PE
// =====================================================================
__global__ __launch_bounds__(256) void rowoffs_k(const int* __restrict__ jarr,
                                                 int* __restrict__ roF,
                                                 int* __restrict__ ro0,
                                                 int* __restrict__ ro1) {
  const int r = blockIdx.x * 256 + threadIdx.x;
  if (r < RST) {
    const int s = r / WWc, p = r % WWc;
    int img, sp;
    if (s < Ntot) { img = 0; sp = s; } else { img = 1; sp = s - Ntot; }
    const int b = sp / Lc;
    const int cell = (img == 0) ? (sp % Lc) : jarr[sp];
    const int oy = cell / 48, ox = cell % 48;
    const int y = oy * 4 - 2 + p / 5, x = ox * 4 - 2 + p % 5;
    int off = -1;
    if (x >= 0 && x < HF && y >= 0 && y < HF)
      off = ((img * 2 + b) * PIX + y * HF + x) * DF;
    roF[r] = off;
  }
  if (r < Ntot * WWc) {
    const int s = r / WWc, p = r % WWc;
    ro0[r] = s * (WWc * DF) + 12 * DF;                 // cen = fd0[:, 12, :]
    ro1[r] = (Ntot + s) * (WWc * DF) + p * DF;         // fd1[:, p, :]
  }
}

__global__ __launch_bounds__(256) void cdesc_k(const float* __restrict__ m0,
                                               const float* __restrict__ m1,
                                               const int* __restrict__ jarr,
                                               float* __restrict__ cd) {
  const int idx = blockIdx.x * 256 + threadIdx.x;
  if (idx >= SEQ * DcK) return;
  const int r = idx / DcK, k = idx % DcK;
  float v;
  if (r < Ntot) v = m0[(size_t)r * DcK + k];
  else {
    const int rr = r - Ntot;
    const int src = (rr / Lc) * Lc + jarr[rr];
    v = m1[(size_t)src * DcK + k];
  }
  cd[idx] = v;
}

__global__ __launch_bounds__(256) void pe_k(float* __restrict__ pe) {
  const int idx = blockIdx.x * 256 + threadIdx.x;
  if (idx >= WWc * DF) return;
  const int p = idx / DF, d = idx % DF;
  const float div = expf(-(float)(d & ~1) * (logf(10000.f) / (float)DF));
  pe[idx] = (d & 1) ? cosf((float)p * div) : sinf((float)p * div);
}

// =====================================================================
// Fused transformer attention layer: 1 block = 1 sequence (T=25, D=128)
// =====================================================================
constexpr int LD1 = 130;   // stride of 128-wide LDS tiles
constexpr int LDH = 258;   // stride of 256-wide h1 tile
constexpr int LDS_FLOATS = 3 * (WWc * LD1) + (WWc * LDH);  // Q,K,V + h1-extension = 16200

// block GEMM: dst(rows<25 x Nout) = [Ag | Al](32xKdim) @ W(Kdim x Nout) (+bias)(relu?)
__device__ __forceinline__ void bgemm(float* __restrict__ dst, int dstride,
                                      const float* __restrict__ Ag,
                                      const float* __restrict__ Al, int alStride, int split,
                                      const float* __restrict__ W, int ldw,
                                      int Kdim, int Nout,
                                      const float* __restrict__ bias, bool relu, int tid) {
  const int wid = tid >> 5, lane = tid & 31, m = lane & 15, h = lane >> 4;
  const int tiles = 2 * (Nout >> 4);
  for (int t = wid; t < tiles; t += 8) {
    const int row0 = (t & 1) << 4;
    const int col0 = (t >> 1) << 4;
    v8f acc = (v8f){0,0,0,0,0,0,0,0};
    for (int kb = 0; kb < Kdim; kb += 4) {
      const int k0 = kb + 2 * h;               // even
      v2f a, b;
      if (k0 < split) {
        a.x = Ag[(row0 + m) * DF + k0];
        a.y = Ag[(row0 + m) * DF + k0 + 1];
      } else {
        a.x = Al[(row0 + m) * alStride + k0 - split];
        a.y = Al[(row0 + m) * alStride + k0 + 1 - split];
      }
      b.x = W[(size_t)k0 * ldw + col0 + m];
      b.y = W[(size_t)(k0 + 1) * ldw + col0 + m];
      acc = wmma4(a, b, acc);
    }
    const int col = col0 + m;
    const float bv = bias ? bias[col] : 0.f;
#pragma unroll
    for (int r = 0; r < 8; ++r) {
      const int row = row0 + r + 8 * h;
      if (row < WWc) {
        float v = acc[r] + bv;
        if (relu) v = fmaxf(v, 0.f);
        dst[row * dstride + col] = v;
      }
    }
  }
}

__device__ __forceinline__ void ln_rows(float* __restrict__ buf, int stride,
                                        const float* __restrict__ s,
                                        const float* __restrict__ b, int tid) {
  const int wid = tid >> 5, lane = tid & 31;
  for (int row = wid; row < WWc; row += 8) {
    float x0 = buf[row * stride + lane];
    float x1 = buf[row * stride + 32 + lane];
    float x2 = buf[row * stride + 64 + lane];
    float x3 = buf[row * stride + 96 + lane];
    float sum = x0 + x1 + x2 + x3;
    for (int o = 16; o; o >>= 1) sum += __shfl_xor(sum, o);
    const float mean = sum * (1.f / 128.f);
    const float d0 = x0 - mean, d1 = x1 - mean, d2 = x2 - mean, d3 = x3 - mean;
    float vs = d0 * d0 + d1 * d1 + d2 * d2 + d3 * d3;
    for (int o = 16; o; o >>= 1) vs += __shfl_xor(vs, o);
    const float rstd = 1.f / sqrtf(vs * (1.f / 128.f) + 1e-5f);
    buf[row * stride + lane]      = d0 * rstd * s[lane]      + b[lane];
    buf[row * stride + 32 + lane] = d1 * rstd * s[32 + lane] + b[32 + lane];
    buf[row * stride + 64 + lane] = d2 * rstd * s[64 + lane] + b[64 + lane];
    buf[row * stride + 96 + lane] = d3 * rstd * s[96 + lane] + b[96 + lane];
  }
}

__global__ __launch_bounds__(256) void attn_k(
    float* __restrict__ state, int xbase, int srcbase,
    const float* __restrict__ wq, const float* __restrict__ wk,
    const float* __restrict__ wv, const float* __restrict__ wm,
    const float* __restrict__ m1w, const float* __restrict__ m1b,
    const float* __restrict__ m2w, const float* __restrict__ m2b,
    const float* __restrict__ n1s, const float* __restrict__ n1b,
    const float* __restrict__ n2s, const float* __restrict__ n2b) {
  __shared__ float smem[LDS_FLOATS];
  float* sQ = smem;                 // q -> att -> h2
  float* sK = smem + WWc * LD1;     // k -> msg(LN)
  float* sV = smem + 2 * WWc * LD1; // v
  float* sH = smem + 2 * WWc * LD1; // h1 (reuses V region + tail)
  const int s = blockIdx.x;
  float* xg = state + (size_t)(xbase + s) * (WWc * DF);
  const float* sg = state + (size_t)(srcbase + s) * (WWc * DF);
  const int tid = threadIdx.x;

  // q = x@wq, k = src@wk, v = src@wv   (A streamed from global)
  bgemm(sQ, LD1, xg, nullptr, 0, 128, wq, DF, 128, 128, nullptr, false, tid);
  bgemm(sK, LD1, sg, nullptr, 0, 128, wk, DF, 128, 128, nullptr, false, tid);
  bgemm(sV, LD1, sg, nullptr, 0, 128, wv, DF, 128, 128, nullptr, false, tid);
  __syncthreads();

  // attention: thread = (head, query-row); att overwrites Q region
  float qr[16]; int hh = 0, qi = 0;
  if (tid < 8 * WWc) {
    hh = tid / WWc; qi = tid % WWc;
#pragma unroll
    for (int d = 0; d < 16; ++d) qr[d] = sQ[qi * LD1 + hh * 16 + d];
  }
  __syncthreads();
  if (tid < 8 * WWc) {
    float sc[WWc]; float mx = -3.4e38f;
    for (int j = 0; j < WWc; ++j) {
      float d = 0.f;
#pragma unroll
      for (int t = 0; t < 16; ++t) d += qr[t] * sK[j * LD1 + hh * 16 + t];
      d *= 0.25f;                       // DH^-0.5
      sc[j] = d; mx = fmaxf(mx, d);
    }
    float sum = 0.f;
    for (int j = 0; j < WWc; ++j) { sc[j] = expf(sc[j] - mx); sum += sc[j]; }
    const float inv = 1.f / sum;
    float out[16];
#pragma unroll
    for (int t = 0; t < 16; ++t) out[t] = 0.f;
    for (int j = 0; j < WWc; ++j) {
      const float a = sc[j] * inv;
#pragma unroll
      for (int t = 0; t < 16; ++t) out[t] += a * sV[j * LD1 + hh * 16 + t];
    }
#pragma unroll
    for (int t = 0; t < 16; ++t) sQ[qi * LD1 + hh * 16 + t] = out[t];
  }
  __syncthreads();

  // msg = att @ wm -> K region, then LN(n1) in place
  bgemm(sK, LD1, nullptr, sQ, LD1, 0, wm, DF, 128, 128, nullptr, false, tid);
  __syncthreads();
  ln_rows(sK, LD1, n1s, n1b, tid);
  __syncthreads();

  // h1 = relu([x | msg] @ m1w + m1b) -> H region
  bgemm(sH, LDH, xg, sK, LD1, 128, m1w, 256, 256, 256, m1b, true, tid);
  __syncthreads();

  // h2 = h1 @ m2w + m2b -> Q region
  bgemm(sQ, LD1, nullptr, sH, LDH, 0, m2w, DF, 256, 128, m2b, false, tid);
  __syncthreads();

  // out = x + LN(h2; n2)
  const int wid = tid >> 5, lane = tid & 31;
  for (int row = wid; row < WWc; row += 8) {
    float x0 = sQ[row * LD1 + lane], x1 = sQ[row * LD1 + 32 + lane];
    float x2 = sQ[row * LD1 + 64 + lane], x3 = sQ[row * LD1 + 96 + lane];
    float sum = x0 + x1 + x2 + x3;
    for (int o = 16; o; o >>= 1) sum += __shfl_xor(sum, o);
    const float mean = sum * (1.f / 128.f);
    const float d0 = x0 - mean, d1 = x1 - mean, d2 = x2 - mean, d3 = x3 - mean;
    float vs = d0 * d0 + d1 * d1 + d2 * d2 + d3 * d3;
    for (int o = 16; o; o >>= 1) vs += __shfl_xor(vs, o);
    const float rstd = 1.f / sqrtf(vs * (1.f / 128.f) + 1e-5f);
    xg[row * DF + lane]      += d0 * rstd * n2s[lane]      + n2b[lane];
    xg[row * DF + 32 + lane] += d1 * rstd * n2s[32 + lane] + n2b[32 + lane];
    xg[row * DF + 64 + lane] += d2 * rstd * n2s[64 + lane] + n2b[64 + lane];
    xg[row * DF + 96 + lane] += d3 * rstd * n2s[96 + lane] + n2b[96 + lane];
  }
}

// =====================================================================
// Final regression + base/mask
// =====================================================================
__global__ __launch_bounds__(256) void final_k(const float* __restrict__ featv,
                                               const float* __restrict__ reg_w,
                                               const float* __restrict__ reg_b,
                                               const int* __restrict__ jarr,
                                               const float* __restrict__ maskf,
                                               float* __restrict__ out) {
  const int s = blockIdx.x * 256 + threadIdx.x;
  if (s >= Ntot) return;
  const float* f = featv + (size_t)s * DF;
  float cx = reg_b[0], cy = reg_b[1];
  for (int k = 0; k < DF; ++k) {
    const float v = f[k];
    cx += v * reg_w[2 * k];
    cy += v * reg_w[2 * k + 1];
  }
  const int jj = jarr[s];
  const float bx = (float)((jj % 48) * 8), by = (float)((jj / 48) * 8);
  const float mk = maskf[s];
  out[2 * s]     = (bx + cx) * mk;
  out[2 * s + 1] = (by + cy) * mk;
}

// =====================================================================
// Host launch
// =====================================================================
extern "C" void kernel_launch(void* const* d_in, const int* in_sizes, int n_in,
                              void* d_out, int out_size, void* d_ws, size_t ws_size,
                              hipStream_t stream) {
  (void)in_sizes; (void)n_in; (void)out_size; (void)ws_size;
  const float* mdesc0 = (const float*)d_in[0];
  const float* mdesc1 = (const float*)d_in[1];
  const float* fine0  = (const float*)d_in[2];
  const float* fine1  = (const float*)d_in[3];
  const float* conv_w = (const float*)d_in[4];
  const float* conv_b = (const float*)d_in[5];
  const float* proj_w = (const float*)d_in[6];
  const float* proj_b = (const float*)d_in[7];
  const float* merge_w= (const float*)d_in[8];
  const float* merge_b= (const float*)d_in[9];
  const float* reg1_w = (const float*)d_in[10];
  const float* reg1_b = (const float*)d_in[11];
  const float* reg2_w = (const float*)d_in[12];
  const float* reg2_b = (const float*)d_in[13];
  const float* reg_w  = (const float*)d_in[14];
  const float* reg_b  = (const float*)d_in[15];
  const float* lt_wq  = (const float*)d_in[16];
  const float* lt_wk  = (const float*)d_in[17];
  const float* lt_wv  = (const float*)d_in[18];
  const float* lt_wm  = (const float*)d_in[19];
  const float* lt_m1w = (const float*)d_in[20];
  const float* lt_m1b = (const float*)d_in[21];
  const float* lt_m2w = (const float*)d_in[22];
  const float* lt_m2b = (const float*)d_in[23];
  const float* lt_n1s = (const float*)d_in[24];
  const float* lt_n1b = (const float*)d_in[25];
  const float* lt_n2s = (const float*)d_in[26];
  const float* lt_n2b = (const float*)d_in[27];
  float* out = (float*)d_out;

  float* w = (float*)d_ws;
  float* state = w + OFF_STATE;
  float* F     = w + OFF_F;
  float* sim   = w + OFF_SIM;
  float* cdesc = w + OFF_CD;
  float* center= w + OFF_CEN;
  float* cmg   = w + OFF_CMG;
  float* pe    = w + OFF_PE;
  float* lrs   = w + OFF_LRS;
  float* lcs   = w + OFF_LCS;
  float* cmax  = w + OFF_CMX;
  float* maskf = w + OFF_MSK;
  int*   jarr  = (int*)(w + OFF_J);
  int*   roF   = (int*)(w + OFF_ROF);
  int*   ro0   = (int*)(w + OFF_RO0);
  int*   ro1   = (int*)(w + OFF_RO1);
  float* feat1 = w + OFF_FEAT1;
  float* featv = w + OFF_FEATV;

  // PE table
  pe_k<<<(WWc * DF + 255) / 256, 256, 0, stream>>>(pe);

  // sim = (1/6553.6) * mdesc0 @ mdesc1^T   (per batch)
  {
    const int Mt = Lc / 16, Nt = Lc / 64;
    gemm_k<false, true><<<dim3((Mt * Nt + 7) / 8, Bc), 256, 0, stream>>>(
        mdesc0, nullptr, DcK, nullptr, nullptr, DcK, mdesc1, DcK, nullptr,
        sim, Lc, Lc, Lc, DcK, 1.0f / 6553.6f, 0, nullptr, nullptr,
        (long)Lc * DcK, (long)Lc * DcK, (long)Lc * Lc);
  }
  rowsum_k<<<Ntot / 8, 256, 0, stream>>>(sim, lrs);
  col_k<<<(Ntot + 255) / 256, 256, 0, stream>>>(sim, lrs, lcs, cmax);
  match_k<<<Ntot / 8, 256, 0, stream>>>(sim, lrs, lcs, cmax, jarr, maskf);

  // 1x1 conv: F[img*2+b] (PIX x 128) = fine[b]^T @ conv_w^T + conv_b
  {
    const int Mt = PIX / 16, Nt = DF / 64;
    gemm_k<true, true><<<dim3((Mt * Nt + 7) / 8, Bc), 256, 0, stream>>>(
        fine0, nullptr, PIX, nullptr, nullptr, DcK, conv_w, DcK, conv_b,
        F, DF, PIX, DF, DcK, 1.0f, 0, nullptr, nullptr,
        (long)DcK * PIX, 0L, (long)PIX * DF);
    gemm_k<true, true><<<dim3((Mt * Nt + 7) / 8, Bc), 256, 0, stream>>>(
        fine1, nullptr, PIX, nullptr, nullptr, DcK, conv_w, DcK, conv_b,
        F + (size_t)2 * PIX * DF, DF, PIX, DF, DcK, 1.0f, 0, nullptr, nullptr,
        (long)DcK * PIX, 0L, (long)PIX * DF);
  }

  rowoffs_k<<<(RST + 255) / 256, 256, 0, stream>>>(jarr, roF, ro0, ro1);
  cdesc_k<<<(SEQ * DcK + 255) / 256, 256, 0, stream>>>(mdesc0, mdesc1, jarr, cdesc);

  // center = cdesc @ proj_w + proj_b
  gemm_k<false, false><<<dim3((SEQ / 16 * (DF / 64) + 7) / 8, 1), 256, 0, stream>>>(
      cdesc, nullptr, DcK, nullptr, nullptr, DcK, proj_w, DF, proj_b,
      center, DF, SEQ, DF, DcK, 1.0f, 0, nullptr, nullptr, 0L, 0L, 0L);

  // cmg = center @ merge_w[128:] + merge_b
  gemm_k<false, false><<<dim3((SEQ / 16 * (DF / 64) + 7) / 8, 1), 256, 0, stream>>>(
      center, nullptr, DF, nullptr, nullptr, DF, merge_w + 128 * DF, DF, merge_b,
      cmg, DF, SEQ, DF, DF, 1.0f, 0, nullptr, nullptr, 0L, 0L, 0L);

  // state = window-gather(F) @ merge_w[:128]  + cmg[s] + pe[p]
  gemm_k<false, false><<<dim3((RST / 16 * (DF / 64) + 7) / 8, 1), 256, 0, stream>>>(
      F, roF, 0, nullptr, nullptr, DF, merge_w, DF, nullptr,
      state, DF, RST, DF, DF, 1.0f, 1, cmg, pe, 0L, 0L, 0L);

  // 4-layer transformer: self, cross, self, cross
  for (int l = 0; l < 4; ++l) {
    const float* wq = lt_wq + (size_t)l * DF * DF;
    const float* wk = lt_wk + (size_t)l * DF * DF;
    const float* wv = lt_wv + (size_t)l * DF * DF;
    const float* wm = lt_wm + (size_t)l * DF * DF;
    const float* m1w = lt_m1w + (size_t)l * 256 * 256;
    const float* m1b = lt_m1b + (size_t)l * 256;
    const float* m2w = lt_m2w + (size_t)l * 256 * DF;
    const float* m2b = lt_m2b + (size_t)l * DF;
    const float* n1s = lt_n1s + (size_t)l * DF;
    const float* n1b = lt_n1b + (size_t)l * DF;
    const float* n2s = lt_n2s + (size_t)l * DF;
    const float* n2b = lt_n2b + (size_t)l * DF;
    if ((l & 1) == 0) {  // self for both streams
      attn_k<<<SEQ, 256, 0, stream>>>(state, 0, 0, wq, wk, wv, wm,
                                      m1w, m1b, m2w, m2b, n1s, n1b, n2s, n2b);
    } else {             // cross: x0 <- attn(x0, x1); then x1 <- attn(x1, new x0)
      attn_k<<<Ntot, 256, 0, stream>>>(state, 0, Ntot, wq, wk, wv, wm,
                                       m1w, m1b, m2w, m2b, n1s, n1b, n2s, n2b);
      attn_k<<<Ntot, 256, 0, stream>>>(state, Ntot, 0, wq, wk, wv, wm,
                                       m1w, m1b, m2w, m2b, n1s, n1b, n2s, n2b);
    }
  }

  // feat1 = [cen | fd1] @ reg1_w + reg1_b
  gemm_k<false, false><<<dim3(((Ntot * WWc) / 16 * (DF / 64) + 7) / 8, 1), 256, 0, stream>>>(
      state, ro0, 0, state, ro1, 128, reg1_w, DF, reg1_b,
      feat1, DF, Ntot * WWc, DF, 2 * DF, 1.0f, 0, nullptr, nullptr, 0L, 0L, 0L);

  // featv = feat1(N x 3200) @ reg2_w + reg2_b
  gemm_k<false, false><<<dim3((Ntot / 16 * (DF / 64) + 7) / 8, 1), 256, 0, stream>>>(
      feat1, nullptr, WWc * DF, nullptr, nullptr, WWc * DF, reg2_w, DF, reg2_b,
      featv, DF, Ntot, DF, WWc * DF, 1.0f, 0, nullptr, nullptr, 0L, 0L, 0L);

  // coords -> (base + coords) * mask
  final_k<<<(Ntot + 255) / 256, 256, 0, stream>>>(featv, reg_w, reg_b, jarr, maskf, out);
}